// MyRNN_89593017795211
// MI455X (gfx1250) — compile-verified
//
#include <hip/hip_runtime.h>
#include <hip/hip_bf16.h>

// Problem constants (from the reference)
#define T_ 8192
#define B_ 512
#define H_ 16
#define OUT_ 5

typedef __attribute__((ext_vector_type(2))) float v2f;
typedef __attribute__((ext_vector_type(8))) float v8f;

__device__ __forceinline__ float fast_sigmoid(float x) {
    // 1 / (1 + e^-x); v_exp_f32 + v_rcp_f32
    return __builtin_amdgcn_rcpf(1.0f + __expf(-x));
}
__device__ __forceinline__ float fast_tanh(float x) {
    // tanh(x) = 2*sigmoid(2x) - 1
    return 2.0f * __builtin_amdgcn_rcpf(1.0f + __expf(-2.0f * x)) - 1.0f;
}
// Uniform-lane broadcast via v_readlane_b32 (result lives in an SGPR; the
// dot-product FMAs can consume it as a scalar source -> no DS round-trip).
__device__ __forceinline__ float lane_bcast(float v, int srcLane) {
    return __int_as_float(__builtin_amdgcn_readlane(__float_as_int(v), srcLane));
}

// ---------------------------------------------------------------------------
// Kernel 1: single-wave GRU recurrence for batch element B-1 only
// (outs[:, -1, :] indexes the batch axis, so only b = B-1 reaches the output;
//  batch elements are independent -> 512x less work than the reference).
// Lanes 0..15 own the r rows, lanes 16..31 the z rows; both halves own the n
// row for their l = lane&15, and after one half-swap shuffle BOTH halves hold
// valid (r, z) and compute identical h_new -> uniform stores, no divergence.
// h[0..15] is kept in SGPRs via v_readlane broadcast each step.
// ---------------------------------------------------------------------------
__global__ __launch_bounds__(32) void gru_seq_kernel(
    const float* __restrict__ x,     // (T, B, 1)
    const float* __restrict__ w_ih,  // (3H, 1)
    const float* __restrict__ w_hh,  // (3H, H)
    const float* __restrict__ b_ih,  // (3H,)
    const float* __restrict__ b_hh,  // (3H,)
    float* __restrict__ hist)        // (T, H) workspace
{
    const int lane = threadIdx.x;        // 0..31
    const int l    = lane & 15;
    const bool lo  = (lane < 16);

    // Gate row A: lanes 0..15 -> rows 0..15 (r); lanes 16..31 -> rows 16..31 (z)
    const int gA = lane;
    // Gate row B: rows 32..47 (n), per l in both halves
    const int gB = 2 * H_ + l;

    float wA[H_], wB[H_];
#pragma unroll
    for (int k = 0; k < H_; ++k) {
        wA[k] = w_hh[gA * H_ + k];
        wB[k] = w_hh[gB * H_ + k];
    }
    const float bhA = b_hh[gA], bhB = b_hh[gB];
    const float aiA = w_ih[gA], biA = b_ih[gA];   // xg coeffs (IN == 1)
    const float aiB = w_ih[gB], biB = b_ih[gB];

    float h[H_];
#pragma unroll
    for (int k = 0; k < H_; ++k) h[k] = 0.0f;

    for (int t0 = 0; t0 < T_; t0 += 32) {
        // Prefetch 32 timesteps of x[., B-1, 0]: one strided load per lane.
        const float xchunk = x[(t0 + lane) * B_ + (B_ - 1)];

        for (int s = 0; s < 32; ++s) {
            // s is wave-uniform -> v_readlane with SGPR lane-select
            const float xv = lane_bcast(xchunk, s);

            // hg dot-products, 4-way split accumulation for ILP
            float aA0 = 0.f, aA1 = 0.f, aA2 = 0.f, aA3 = 0.f;
            float aB0 = 0.f, aB1 = 0.f, aB2 = 0.f, aB3 = 0.f;
#pragma unroll
            for (int k = 0; k < H_; k += 4) {
                aA0 = __builtin_fmaf(wA[k + 0], h[k + 0], aA0);
                aA1 = __builtin_fmaf(wA[k + 1], h[k + 1], aA1);
                aA2 = __builtin_fmaf(wA[k + 2], h[k + 2], aA2);
                aA3 = __builtin_fmaf(wA[k + 3], h[k + 3], aA3);
                aB0 = __builtin_fmaf(wB[k + 0], h[k + 0], aB0);
                aB1 = __builtin_fmaf(wB[k + 1], h[k + 1], aB1);
                aB2 = __builtin_fmaf(wB[k + 2], h[k + 2], aB2);
                aB3 = __builtin_fmaf(wB[k + 3], h[k + 3], aB3);
            }
            const float hgA = bhA + (aA0 + aA1) + (aA2 + aA3);
            const float hgB = bhB + (aB0 + aB1) + (aB2 + aB3);

            // Low lanes: sA == r(l). High lanes: sA == z(l).
            const float sA = fast_sigmoid(__builtin_fmaf(xv, aiA, biA) + hgA);
            const float sw = __shfl_xor(sA, 16, 32);   // the other gate
            const float r  = lo ? sA : sw;
            const float z  = lo ? sw : sA;
            const float n  = fast_tanh(__builtin_fmaf(xv, aiB, biB) + r * hgB);
            const float hn = (1.0f - z) * n + z * h[l];

            // All 32 lanes hold valid hn (duplicated across halves):
            // uniform store, lanes 16..31 rewrite the same values.
            hist[(t0 + s) * H_ + l] = hn;

            // Re-broadcast h into SGPRs (constant lane -> v_readlane imm)
#pragma unroll
            for (int k = 0; k < H_; ++k) h[k] = lane_bcast(hn, k);
        }
    }
}

// ---------------------------------------------------------------------------
// Kernel 2: out = hist(8192x16) @ fc_w.T(16x5) + fc_b via V_WMMA_F32_16X16X4_F32.
// One wave per 16-row tile; K=16 done as 4 accumulating k=4 WMMAs.
// 32-bit A layout: lanes 0..15 hold M=lane {K=0,1} in {v0,v1};
//                  lanes 16..31 hold M=lane-16 {K=2,3}.  B mirrors with N=lane&15.
// D layout: VGPR v, lane l -> M = v + 8*(l>=16), N = l&15.
// B loads use a clamped (always in-bounds) row index + cndmask zeroing so the
// compiler emits straight-line loads instead of EXEC-masked branch regions.
// ---------------------------------------------------------------------------
__global__ __launch_bounds__(32) void fc_wmma_kernel(
    const float* __restrict__ hist,  // (T, H)
    const float* __restrict__ fc_w,  // (OUT, H) row-major
    const float* __restrict__ fc_b,  // (OUT,)
    float* __restrict__ out)         // (T, OUT)
{
    const int tile   = blockIdx.x;        // T/16 tiles
    const int lane   = threadIdx.x;       // 0..31
    const int nl     = lane & 15;
    const int hiHalf = lane >> 4;         // 0: K 0,1 ; 1: K 2,3

    const bool valid  = (nl < OUT_);
    const int  nclamp = valid ? nl : 0;   // safe row of fc_w
    const int  m      = tile * 16 + nl;   // A-matrix row for this lane

    // Preload B (fc_w.T zero-padded to 16 cols): unconditional clamped loads,
    // then select-to-zero (v_cndmask) for the padding columns.
    v2f b[4];
#pragma unroll
    for (int kc = 0; kc < 4; ++kc) {
        const int k0 = kc * 4 + 2 * hiHalf;
        const float b0 = fc_w[nclamp * H_ + k0 + 0];
        const float b1 = fc_w[nclamp * H_ + k0 + 1];
        b[kc].x = valid ? b0 : 0.0f;
        b[kc].y = valid ? b1 : 0.0f;
    }

    v8f acc = {};                         // C = 0
#pragma unroll
    for (int kc = 0; kc < 4; ++kc) {
        const int k0 = kc * 4 + 2 * hiHalf;
        v2f a;
        a.x = hist[m * H_ + k0 + 0];
        a.y = hist[m * H_ + k0 + 1];
        acc = __builtin_amdgcn_wmma_f32_16x16x4_f32(
            /*neg_a=*/false, a, /*neg_b=*/false, b[kc],
            /*c_mod=*/(short)0, acc, /*reuse_a=*/false, /*reuse_b=*/false);
    }

    if (valid) {
        const float bias = fc_b[nl];
#pragma unroll
        for (int v = 0; v < 8; ++v) {
            const int row = tile * 16 + v + 8 * hiHalf;
            out[row * OUT_ + nl] = acc[v] + bias;
        }
    }
}

extern "C" void kernel_launch(void* const* d_in, const int* in_sizes, int n_in,
                              void* d_out, int out_size, void* d_ws, size_t ws_size,
                              hipStream_t stream) {
    const float* x    = (const float*)d_in[0];
    const float* w_ih = (const float*)d_in[1];
    const float* w_hh = (const float*)d_in[2];
    const float* b_ih = (const float*)d_in[3];
    const float* b_hh = (const float*)d_in[4];
    const float* fc_w = (const float*)d_in[5];
    const float* fc_b = (const float*)d_in[6];
    float* out  = (float*)d_out;
    float* hist = (float*)d_ws;           // T*H floats = 512 KB

    gru_seq_kernel<<<1, 32, 0, stream>>>(x, w_ih, w_hh, b_ih, b_hh, hist);
    fc_wmma_kernel<<<T_ / 16, 32, 0, stream>>>(hist, fc_w, fc_b, out);
}